// SSIM3D_47347719471117
// MI455X (gfx1250) — compile-verified
//
#include <hip/hip_runtime.h>

// ---------------------------------------------------------------------------
// 3D SSIM (window=7, sigma=1.5) fused tile kernel for gfx1250 (MI455X).
// Separable Gaussian blur of 5 fields as banded matmuls (V_WMMA_F32_16X16X4_F32),
// TDM (tensor_load_to_lds) halo staging for interior tiles, skewed LDS layouts
// for conflict-free DS traffic, deterministic two-stage reduction.
// ---------------------------------------------------------------------------

typedef __attribute__((ext_vector_type(2))) float v2f;
typedef __attribute__((ext_vector_type(8))) float v8f;
typedef __attribute__((ext_vector_type(4))) unsigned int v4u;
typedef __attribute__((ext_vector_type(4))) int v4i;
typedef __attribute__((ext_vector_type(8))) int v8i;

#define NVOX 160
#define TILE 16
#define HALO 22            // TILE + 6
#define XPITCH 24          // xh/yh w-pitch: matches TDM contiguous 24-float rows
#define B1PITCH 28         // buf1 h-pitch  (gcd(28,64)=4 -> conflict-free b64 loads)
#define B2SLAB 452         // buf2 per-h slab stride = 16*28 + 4 (skew, ==4 mod 64)
#define RSLAB 260          // res  per-d stride      = 256 + 4   (skew, ==4 mod 64)
#define RFLD (TILE * RSLAB)  // 4160 floats per field
#define SSIM_C1 1.0e-4f
#define SSIM_C2 9.0e-4f

#if defined(__gfx1250__) && __has_builtin(__builtin_amdgcn_tensor_load_to_lds) && \
    __has_builtin(__builtin_amdgcn_s_wait_tensorcnt)
#define HAVE_TDM 1
#else
#define HAVE_TDM 0
#endif

// Normalized 1D Gaussian (k=7, sigma=1.5); the reference 3D kernel is exactly
// the outer product of this vector.
__device__ __forceinline__ float gauss_tap(int t) {
  const float W7[7] = {0.03663285f, 0.11128076f, 0.21674532f, 0.27068212f,
                       0.21674532f, 0.11128076f, 0.03663285f};
  float r = 0.0f;
#pragma unroll
  for (int q = 0; q < 7; ++q) r = (t == q) ? W7[q] : r;
  return r;
}

// Banded kernel-matrix entry K[i][c] = w[i-c] for 0 <= i-c <= 6, else 0.
__device__ __forceinline__ float bandw(int i, int c) { return gauss_tap(i - c); }

template <int F>
__device__ __forceinline__ float fieldvalT(float xv, float yv) {
  if constexpr (F == 0) return xv;
  else if constexpr (F == 1) return yv;
  else if constexpr (F == 2) return xv * xv;
  else if constexpr (F == 3) return yv * yv;
  else return xv * yv;
}

// 16-line x 16-output 1D conv via 6x V_WMMA_F32_16X16X4_F32.
// A[m][k] = A + m*sM + k  (K axis contiguous, banded/zero-padded to k<24).
__device__ __forceinline__ v8f conv16k(const float* __restrict__ A, int sM,
                                       int lane, const v2f* __restrict__ bw) {
  const int m = lane & 15;
  const int kof = (lane >> 4) << 1;  // lanes 0-15: K=0,1 ; lanes 16-31: K=2,3
  const float* ap = A + m * sM;
  v8f c = {};
#pragma unroll
  for (int ch = 0; ch < 6; ++ch) {
    v2f a = *(const v2f*)(ap + ch * 4 + kof);  // 8B-aligned ds_load_b64
    c = __builtin_amdgcn_wmma_f32_16x16x4_f32(false, a, false, bw[ch],
                                              (short)0, c, false, false);
  }
  return c;
}

// One field, three banded-matmul passes; layouts keep the conv axis contiguous:
//   xh/yh : [22 d][22 h][24 w]     buf1 : [22 d][16 w][28 h-pitch]
//   buf2  : [16 h][16 w -> slab 452][d]   res : [16 d -> 260][16 h][16 w]
template <int F>
__device__ __forceinline__ void blur_field(const float* __restrict__ xh,
                                           const float* __restrict__ yh,
                                           float* __restrict__ buf1,
                                           float* __restrict__ buf2,
                                           float* __restrict__ resf,
                                           const v2f* __restrict__ bw,
                                           int lane, int wvu) {
  // ---- Pass W (conv along w). Two h-groups; overlap rows h=6..15 are
  // bitwise-identical in both groups, so unconditional stores are benign.
#pragma unroll
  for (int g = 0; g < 2; ++g) {
    const int hbase = g * 6;  // group0: h 0..15, group1: h 6..21
    for (int d = wvu; d < HALO; d += 8) {
      const int m = lane & 15, kof = (lane >> 4) << 1;
      const float* xp = xh + (d * HALO + hbase + m) * XPITCH;
      const float* yp = yh + (d * HALO + hbase + m) * XPITCH;
      v8f c = {};
#pragma unroll
      for (int ch = 0; ch < 6; ++ch) {
        const int k = ch * 4 + kof;
        v2f ax = *(const v2f*)(xp + k);
        v2f ay = *(const v2f*)(yp + k);  // DCE'd for F==0
        v2f a;
        a.x = fieldvalT<F>(ax.x, ay.x);
        a.y = fieldvalT<F>(ax.y, ay.y);
        c = __builtin_amdgcn_wmma_f32_16x16x4_f32(false, a, false, bw[ch],
                                                  (short)0, c, false, false);
      }
      const int n = lane & 15, mb = (lane >> 4) << 3;
      float* dp = buf1 + (d * TILE + n) * B1PITCH + hbase + mb;  // [d][w=n][h]
#pragma unroll
      for (int r = 0; r < 8; ++r) dp[r] = c[r];
    }
  }
  __syncthreads();

  // ---- Pass H (conv along h): M = w lines, K = h (contiguous in buf1).
  for (int d = wvu; d < HALO; d += 8) {
    v8f c = conv16k(buf1 + d * TILE * B1PITCH, B1PITCH, lane, bw);
    const int n = lane & 15, mb = (lane >> 4) << 3;
    float* dp = buf2 + n * B2SLAB + mb * B1PITCH + d;  // [h=n][w=mm][d]
#pragma unroll
    for (int r = 0; r < 8; ++r) dp[r * B1PITCH] = c[r];
  }
  __syncthreads();

  // ---- Pass D (conv along d): M = w lines, K = d (contiguous in buf2).
  for (int h = wvu; h < TILE; h += 8) {
    v8f c = conv16k(buf2 + h * B2SLAB, B1PITCH, lane, bw);
    const int n = lane & 15, mb = (lane >> 4) << 3;
    float* dp = resf + n * RSLAB + h * TILE + mb;  // res[d=n][h][w]
#pragma unroll
    for (int r = 0; r < 8; ++r) dp[r] = c[r];
  }
  __syncthreads();
}

#if HAVE_TDM
// DMA a fully in-bounds 24x22x22 f32 tile (rows of 24 = the LDS pitch) from
// global into LDS via the Tensor Data Mover. Descriptor per CDNA5 ISA.
__device__ __forceinline__ void tdm_load_tile(const float* gbase,
                                              unsigned ldsByteAddr) {
  const unsigned long long ga = (unsigned long long)(size_t)gbase;
  v4u g0;
  g0.x = 1u;                                   // count=1, user descriptor
  g0.y = ldsByteAddr;                          // lds_addr [63:32]
  g0.z = (unsigned)(ga & 0xFFFFFFFFu);         // global_addr low
  g0.w = (unsigned)((ga >> 32) & 0x01FFFFFFu)  // global_addr[56:32]
         | 0x80000000u;                        // type=2 ("image")
  v8i g1;
  g1[0] = 0x00020000;                          // data_size = 4B
  g1[1] = (int)((NVOX & 0xFFFF) << 16);        // tensor_dim0[15:0]
  g1[2] = (int)((NVOX & 0xFFFF) << 16);        // dim0[31:16]=0 | dim1[15:0]
  g1[3] = (int)(XPITCH << 16);                 // dim1[31:16]=0 | tile_dim0=24
  g1[4] = HALO | (HALO << 16);                 // tile_dim1=22 | tile_dim2=22
  g1[5] = NVOX;                                // dim0_stride low32 = 160
  g1[6] = (int)(((NVOX * NVOX) & 0xFFFF) << 16);  // d0s hi=0 | dim1_stride lo16
  g1[7] = (NVOX * NVOX) >> 16;                 // dim1_stride[47:16] = 0
  v4i g2;
  g2.x = NVOX;                                 // tensor_dim2
  g2.y = 0; g2.z = 0; g2.w = 0;                // tile_dim3 unused
  v4i g3 = {0, 0, 0, 0};
#if __has_include(<hip/amd_detail/amd_gfx1250_TDM.h>)
  v8i g4 = {0, 0, 0, 0, 0, 0, 0, 0};           // therock 6-arg form
  __builtin_amdgcn_tensor_load_to_lds(g0, g1, g2, g3, g4, 0);
#else
  __builtin_amdgcn_tensor_load_to_lds(g0, g1, g2, g3, 0);  // ROCm 7.2 5-arg
#endif
}
#endif

__global__ void __launch_bounds__(256)
ssim3d_tile_kernel(const float* __restrict__ x, const float* __restrict__ y,
                   float* __restrict__ partial) {
  extern __shared__ float smem[];
  float* xh   = smem;                          // [22][22][24]
  float* yh   = xh + HALO * HALO * XPITCH;     // [22][22][24]
  float* buf1 = yh + HALO * HALO * XPITCH;     // [22][16][28]
  float* buf2 = buf1 + HALO * TILE * B1PITCH;  // 16 * 452
  float* res  = buf2 + TILE * B2SLAB;          // [5][16*260]

  const int tid  = threadIdx.x;
  const int lane = tid & 31;
  const int wvu  = __builtin_amdgcn_readfirstlane(tid >> 5);  // scalar wave id

  const int w0 = blockIdx.x * TILE;
  const int h0 = blockIdx.y * TILE;
  const int b  = blockIdx.z / 10;
  const int dt = blockIdx.z % 10;
  const int d0 = dt * TILE;
  const size_t baseIn = (size_t)b * NVOX * NVOX * NVOX;

  // Zero conv-axis pads of buf1 (h=22,23) and buf2 (d=22,23) once; the passes
  // never write them.
  for (int i = tid; i < HALO * TILE * 2; i += 256)
    buf1[(i >> 1) * B1PITCH + HALO + (i & 1)] = 0.0f;
  for (int i = tid; i < TILE * TILE * 2; i += 256) {
    const int hw = i >> 1;
    buf2[(hw >> 4) * B2SLAB + (hw & 15) * B1PITCH + HALO + (i & 1)] = 0.0f;
  }

  // ---- Stage x/y halos into LDS.
  auto scalar_halo = [&]() {
    for (int i = tid; i < HALO * HALO * 2; i += 256) {  // w-pad columns 22,23
      int ld = i / 44, lh = (i % 44) >> 1, lw = HALO + (i & 1);
      xh[(ld * HALO + lh) * XPITCH + lw] = 0.0f;
      yh[(ld * HALO + lh) * XPITCH + lw] = 0.0f;
    }
    for (int i = tid; i < HALO * HALO * HALO; i += 256) {
      int lw = i % HALO, lh = (i / HALO) % HALO, ld = i / (HALO * HALO);
      int gw = w0 - 3 + lw, gh = h0 - 3 + lh, gd = d0 - 3 + ld;
      float xv = 0.0f, yv = 0.0f;
      if ((unsigned)gw < NVOX && (unsigned)gh < NVOX && (unsigned)gd < NVOX) {
        size_t off = baseIn + ((size_t)gd * NVOX + gh) * NVOX + gw;
        xv = x[off];
        yv = y[off];
      }
      xh[(ld * HALO + lh) * XPITCH + lw] = xv;
      yh[(ld * HALO + lh) * XPITCH + lw] = yv;
    }
  };

#if HAVE_TDM
  const bool interior = (blockIdx.x >= 1 && blockIdx.x <= 8 &&
                         blockIdx.y >= 1 && blockIdx.y <= 8 &&
                         dt >= 1 && dt <= 8);
  if (interior) {
    // 24-wide rows are fully in-bounds for interior tiles; the two extra
    // columns land in the w-pad and meet only zero band weights.
    const float* gx = x + baseIn + ((size_t)(d0 - 3) * NVOX + (h0 - 3)) * NVOX + (w0 - 3);
    const float* gy = y + baseIn + ((size_t)(d0 - 3) * NVOX + (h0 - 3)) * NVOX + (w0 - 3);
    if (wvu == 0) tdm_load_tile(gx, (unsigned)(size_t)(void*)xh);
    if (wvu == 1) tdm_load_tile(gy, (unsigned)(size_t)(void*)yh);
  } else {
    scalar_halo();
  }
#else
  scalar_halo();
#endif

  // Per-lane banded B chunks (overlaps with the in-flight TDM transfers):
  // B[k][n] with k = 4*ch + kof + {0,1}, n = lane&15.
  v2f bw[6];
  {
    const int n = lane & 15;
    const int kof = (lane >> 4) << 1;
#pragma unroll
    for (int ch = 0; ch < 6; ++ch) {
      const int i0 = ch * 4 + kof;
      bw[ch].x = bandw(i0, n);
      bw[ch].y = bandw(i0 + 1, n);
    }
  }
#if HAVE_TDM
  if (interior) __builtin_amdgcn_s_wait_tensorcnt(0);
#endif
  __syncthreads();

  blur_field<0>(xh, yh, buf1, buf2, res + 0 * RFLD, bw, lane, wvu);
  blur_field<1>(xh, yh, buf1, buf2, res + 1 * RFLD, bw, lane, wvu);
  blur_field<2>(xh, yh, buf1, buf2, res + 2 * RFLD, bw, lane, wvu);
  blur_field<3>(xh, yh, buf1, buf2, res + 3 * RFLD, bw, lane, wvu);
  blur_field<4>(xh, yh, buf1, buf2, res + 4 * RFLD, bw, lane, wvu);

  // ---- Pointwise SSIM + deterministic block reduction.
  float sum = 0.0f;
  for (int v = tid; v < 4096; v += 256) {
    const int ix = (v >> 8) * RSLAB + (v & 255);  // skewed res indexing
    const float mx  = res[ix];
    const float my  = res[RFLD + ix];
    const float ex2 = res[2 * RFLD + ix];
    const float ey2 = res[3 * RFLD + ix];
    const float exy = res[4 * RFLD + ix];
    const float sx  = fmaxf(ex2 - mx * mx, 0.0f);
    const float sy  = fmaxf(ey2 - my * my, 0.0f);
    const float sxy = exy - mx * my;
    const float num = (2.0f * mx * my + SSIM_C1) * (2.0f * sxy + SSIM_C2);
    float den = (mx * mx + my * my + SSIM_C1) * (sx + sy + SSIM_C2);
    den = fmaxf(den, 1e-8f);
    sum += num / den;
  }
  __syncthreads();
  float* red = smem;  // safe: all LDS consumers are behind the barrier
  red[tid] = sum;
  __syncthreads();
#pragma unroll
  for (int s = 128; s > 0; s >>= 1) {
    if (tid < s) red[tid] += red[tid + s];
    __syncthreads();
  }
  if (tid == 0)
    partial[(blockIdx.z * 10 + blockIdx.y) * 10 + blockIdx.x] = red[0];
}

__global__ void __launch_bounds__(256)
ssim3d_finalize(const float* __restrict__ partial, float* __restrict__ out) {
  __shared__ float red[256];
  float s = 0.0f;
  for (int i = threadIdx.x; i < 2000; i += 256) s += partial[i];
  red[threadIdx.x] = s;
  __syncthreads();
#pragma unroll
  for (int k = 128; k > 0; k >>= 1) {
    if (threadIdx.x < k) red[threadIdx.x] += red[threadIdx.x + k];
    __syncthreads();
  }
  if (threadIdx.x == 0)
    out[0] = 1.0f - red[0] * (1.0f / (2.0f * 160.0f * 160.0f * 160.0f));
}

extern "C" void kernel_launch(void* const* d_in, const int* in_sizes, int n_in,
                              void* d_out, int out_size, void* d_ws, size_t ws_size,
                              hipStream_t stream) {
  const float* x = (const float*)d_in[0];
  const float* y = (const float*)d_in[1];
  float* partial = (float*)d_ws;   // 2000 f32 partials
  float* out     = (float*)d_out;  // single f32

  const size_t smemBytes =
      (size_t)(2 * HALO * HALO * XPITCH + HALO * TILE * B1PITCH + TILE * B2SLAB +
               5 * RFLD) * sizeof(float);  // ~239 KB < 320 KB/WGP

  dim3 grid(10, 10, 20);  // 10x10 (w,h) tiles, z = batch*10 + d-tile
  dim3 block(256);        // 8 wave32
  hipLaunchKernelGGL(ssim3d_tile_kernel, grid, block, smemBytes, stream,
                     x, y, partial);
  hipLaunchKernelGGL(ssim3d_finalize, dim3(1), dim3(256), 0, stream,
                     partial, out);
}